// RNN_8392366096600
// MI455X (gfx1250) — compile-verified
//
#include <hip/hip_runtime.h>
#include <hip/hip_bf16.h>
#include <math.h>

// ---------------------------------------------------------------------------
// LSTM stack (T=4096, B=1, IN=40, H=512, L=3) + MLP head, for MI455X (gfx1250)
//  * input GEMMs: V_WMMA_F32_16X16X4_F32 (fp32 tensor op, exact precision)
//  * recurrence:  persistent 32-WG kernel; W_hh slice staged ONCE into LDS
//                 via GLOBAL_LOAD_ASYNC_TO_LDS_B128 (ASYNCcnt) and reused for
//                 all 4096 steps; per-step grid barrier via agent atomics.
// ---------------------------------------------------------------------------

#define Tdim 4096
#define INdim 40
#define Hdim 512
#define G4 2048            // 4*H
#define NWG 32             // workgroups in persistent recurrence kernel
#define UNITS_PER_WG 16    // 512 / 32
#define ROWS_PER_WG 64     // 4 gates * 16 units
#define SLICE_F 132        // 128 floats used + 4 pad (bank de-conflict, 16B ok)
#define WREC_LDS_BYTES (256 * SLICE_F * 4)   // 135168 B < 320 KB WGP LDS

typedef float v2f __attribute__((ext_vector_type(2)));
typedef float v8f __attribute__((ext_vector_type(8)));

// ---------------------------------------------------------------------------
// xw[t][n] = sum_k X[t][k] * Wih[n][k] + b_ih[n] + b_hh[n]
// X: [T, K] row-major, Wih: [2048, K] row-major, xw: [T, 2048]
// One wave per 16x16 output tile, fp32 WMMA over K in steps of 4.
// ---------------------------------------------------------------------------
__global__ __launch_bounds__(128)
void gemm_xw_kernel(const float* __restrict__ X,
                    const float* __restrict__ Wih,
                    const float* __restrict__ b_ih,
                    const float* __restrict__ b_hh,
                    float* __restrict__ xw,
                    int K) {
    const int lane = threadIdx.x & 31;
    const int wv   = threadIdx.x >> 5;            // wave in block: 0..3
    const int n0   = (blockIdx.x * 4 + wv) * 16;  // output column tile base
    const int t0   = blockIdx.y * 16;             // output row tile base
    const int mcol = lane & 15;                   // M for A / N for B / N for C
    const int hi   = lane >> 4;
    const int koff = hi << 1;                     // lanes>=16 carry K+2,K+3

    const float* Arow = X   + (size_t)(t0 + mcol) * K + koff;
    const float* Brow = Wih + (size_t)(n0 + mcol) * K + koff;

    v8f acc = {};
    for (int k0 = 0; k0 < K; k0 += 4) {
        v2f a, b;
        a.x = Arow[k0];     // A[m][k0+koff]
        a.y = Arow[k0 + 1]; // A[m][k0+koff+1]
        b.x = Brow[k0];     // B[k0+koff][n] = Wih[n][k0+koff]
        b.y = Brow[k0 + 1];
        acc = __builtin_amdgcn_wmma_f32_16x16x4_f32(
            /*neg_a=*/false, a, /*neg_b=*/false, b,
            /*c_mod=*/(short)0, acc, /*reuse_a=*/false, /*reuse_b=*/false);
    }

    const float bias = b_ih[n0 + mcol] + b_hh[n0 + mcol];
#pragma unroll
    for (int v = 0; v < 8; ++v) {
        const int m = v + 8 * hi;   // C layout: VGPR v -> M=v (lanes 0-15) / v+8
        xw[(size_t)(t0 + m) * G4 + n0 + mcol] = acc[v] + bias;
    }
}

// ---------------------------------------------------------------------------
// gfx1250 async global -> LDS copy (16 B per lane), drained via ASYNCcnt.
// ---------------------------------------------------------------------------
__device__ __forceinline__ void async_copy16_to_lds(unsigned lds_byte_off,
                                                    const void* gaddr) {
    asm volatile("global_load_async_to_lds_b128 %0, %1, off"
                 :: "v"(lds_byte_off), "v"(gaddr)
                 : "memory");
}

__device__ __forceinline__ void wait_asynccnt0() {
    asm volatile("s_wait_asynccnt 0" ::: "memory");
}

__device__ __forceinline__ float sigmoidf_(float x) {
    return 1.0f / (1.0f + __expf(-x));
}

// ---------------------------------------------------------------------------
// Persistent recurrence kernel: 32 WGs x 256 threads. WG `wg` owns hidden
// units [wg*16, wg*16+16) and their 4 gate rows each (64 rows of W_hh).
// Thread tid owns row (tid>>2), K-slice (tid&3)*128..+128, held in LDS for
// the whole sequence. Double-buffered h in global memory + grid barrier.
// ---------------------------------------------------------------------------
__global__ __launch_bounds__(256)
void lstm_rec_kernel(const float* __restrict__ xw,    // [T, 2048]
                     const float* __restrict__ Whh,   // [2048, 512]
                     float* h_buf,                    // [2, 512] (zero-init)
                     unsigned int* sync_cnt,          // zero-init
                     float* __restrict__ hout) {      // [T, 512]
    __shared__ float s_h[Hdim];
    __shared__ float s_part[256];
    __shared__ float s_g[ROWS_PER_WG];
    extern __shared__ __align__(16) float s_w[];      // 256 slices * SLICE_F

    const int tid = threadIdx.x;
    const int wg  = blockIdx.x;              // 0..31
    const int row_local = tid >> 2;          // 0..63
    const int part      = tid & 3;           // 0..3 (128-wide K slice each)
    const int gate   = row_local >> 4;       // 0..3 (i,f,g,o)
    const int ulocal = row_local & 15;
    const int grow   = gate * Hdim + wg * UNITS_PER_WG + ulocal; // global row

    // ---- stage this WG's 64x512 W_hh slice into LDS once (async DMA) ----
    {
        const float* gsrc = Whh + (size_t)grow * Hdim + part * 128;
        const unsigned lds_base =
            (unsigned)(size_t)(void*)s_w + (unsigned)tid * (SLICE_F * 4u);
#pragma unroll
        for (int j = 0; j < 32; ++j) {       // 32 x 16B = 512B = 128 floats
            async_copy16_to_lds(lds_base + (unsigned)j * 16u, gsrc + j * 4);
        }
        wait_asynccnt0();
    }
    __syncthreads();

    const float* wl = s_w + (size_t)tid * SLICE_F;   // this thread's weights
    float c = 0.0f;                          // cell state (threads 0..15 only)

    for (int t = 0; t < Tdim; ++t) {
        // ---- load h(t-1) from global double buffer into LDS ----
        const float* hrd = h_buf + (t & 1) * Hdim;
        for (int i = tid; i < Hdim; i += 256) s_h[i] = hrd[i];
        __syncthreads();

        // ---- partial dot: 128 LDS-resident MACs per thread ----
        float s = 0.0f;
        const float* hp = s_h + part * 128;
#pragma unroll 8
        for (int k = 0; k < 128; ++k) s = fmaf(wl[k], hp[k], s);
        s_part[tid] = s;
        __syncthreads();

        if (part == 0) {
            float g = xw[(size_t)t * G4 + grow]
                    + s_part[tid] + s_part[tid + 1]
                    + s_part[tid + 2] + s_part[tid + 3];
            s_g[row_local] = g;
        }
        __syncthreads();

        // ---- gate combine + cell update for our 16 units ----
        if (tid < UNITS_PER_WG) {
            const float gi = s_g[tid];
            const float gf = s_g[16 + tid];
            const float gg = s_g[32 + tid];
            const float go = s_g[48 + tid];
            const float i_ = sigmoidf_(gi);
            const float f_ = sigmoidf_(gf);
            const float g_ = tanhf(gg);
            const float o_ = sigmoidf_(go);
            c = fmaf(f_, c, i_ * g_);
            const float hn = o_ * tanhf(c);
            const int u = wg * UNITS_PER_WG + tid;
            h_buf[((t + 1) & 1) * Hdim + u] = hn;
            hout[(size_t)t * Hdim + u] = hn;
            __threadfence();
        }
        __syncthreads();

        // ---- grid barrier (monotonic counter, release/acquire) ----
        if (tid == 0) {
            __hip_atomic_fetch_add(sync_cnt, 1u, __ATOMIC_RELEASE,
                                   __HIP_MEMORY_SCOPE_AGENT);
            const unsigned int target = (unsigned int)NWG * (unsigned int)(t + 1);
            while (__hip_atomic_load(sync_cnt, __ATOMIC_ACQUIRE,
                                     __HIP_MEMORY_SCOPE_AGENT) < target) {}
        }
        __syncthreads();
    }
}

// ---------------------------------------------------------------------------
// Head: per timestep, a = tanh(W1 h + b1) [26]; logits = W2 a + b2 [2]; softmax
// One wave per timestep.
// ---------------------------------------------------------------------------
__global__ __launch_bounds__(32)
void mlp_head_kernel(const float* __restrict__ h,   // [T, 512]
                     const float* __restrict__ W1,  // [26, 512]
                     const float* __restrict__ b1,  // [26]
                     const float* __restrict__ W2,  // [2, 26]
                     const float* __restrict__ b2,  // [2]
                     float* __restrict__ out) {     // [T, 2]
    __shared__ float s_h[Hdim];
    __shared__ float s_a[26];
    const int t    = blockIdx.x;
    const int lane = threadIdx.x;

    for (int i = lane; i < Hdim; i += 32) s_h[i] = h[(size_t)t * Hdim + i];
    __syncthreads();

    if (lane < 26) {
        float s = b1[lane];
        const float* w = W1 + (size_t)lane * Hdim;
        for (int k = 0; k < Hdim; ++k) s = fmaf(w[k], s_h[k], s);
        s_a[lane] = tanhf(s);
    }
    __syncthreads();

    if (lane == 0) {
        float l0 = b2[0], l1 = b2[1];
        for (int j = 0; j < 26; ++j) {
            l0 = fmaf(W2[j], s_a[j], l0);
            l1 = fmaf(W2[26 + j], s_a[j], l1);
        }
        const float m  = fmaxf(l0, l1);
        const float e0 = __expf(l0 - m);
        const float e1 = __expf(l1 - m);
        const float inv = 1.0f / (e0 + e1);
        out[(size_t)t * 2 + 0] = e0 * inv;
        out[(size_t)t * 2 + 1] = e1 * inv;
    }
}

// ---------------------------------------------------------------------------
// Launch sequence: per layer -> GEMM (WMMA), reset sync region, recurrence.
// Then MLP head. Everything on `stream`, graph-capture safe.
// ---------------------------------------------------------------------------
extern "C" void kernel_launch(void* const* d_in, const int* in_sizes, int n_in,
                              void* d_out, int out_size, void* d_ws, size_t ws_size,
                              hipStream_t stream) {
    (void)in_sizes; (void)n_in; (void)out_size; (void)ws_size;

    const float* x = (const float*)d_in[0];
    const float* W_ih[3] = { (const float*)d_in[1], (const float*)d_in[5], (const float*)d_in[9]  };
    const float* W_hh[3] = { (const float*)d_in[2], (const float*)d_in[6], (const float*)d_in[10] };
    const float* b_ih[3] = { (const float*)d_in[3], (const float*)d_in[7], (const float*)d_in[11] };
    const float* b_hh[3] = { (const float*)d_in[4], (const float*)d_in[8], (const float*)d_in[12] };
    const float* W1 = (const float*)d_in[13];
    const float* b1 = (const float*)d_in[14];
    const float* W2 = (const float*)d_in[15];
    const float* b2 = (const float*)d_in[16];
    float* out = (float*)d_out;

    char* ws = (char*)d_ws;
    const size_t XW_BYTES = (size_t)Tdim * G4 * sizeof(float);    // 32 MB
    const size_t H_BYTES  = (size_t)Tdim * Hdim * sizeof(float);  // 8 MB
    float* xw   = (float*)(ws);
    float* hA   = (float*)(ws + XW_BYTES);
    float* hB   = (float*)(ws + XW_BYTES + H_BYTES);
    float* hbuf = (float*)(ws + XW_BYTES + 2 * H_BYTES);          // 2*512 floats
    unsigned int* cnt = (unsigned int*)(ws + XW_BYTES + 2 * H_BYTES
                                        + 2 * Hdim * sizeof(float));

    const float* layer_in = x;
    float* layer_out = hA;
    int K = INdim;

    for (int l = 0; l < 3; ++l) {
        // xw = layer_in @ W_ih^T + b_ih + b_hh   (WMMA fp32)
        dim3 ggrid(G4 / (16 * 4), Tdim / 16);   // (32, 256)
        gemm_xw_kernel<<<ggrid, 128, 0, stream>>>(layer_in, W_ih[l],
                                                  b_ih[l], b_hh[l], xw, K);

        // reset h double-buffer (h0 = 0) and grid-barrier counter
        hipMemsetAsync(hbuf, 0, 2 * Hdim * sizeof(float) + sizeof(unsigned int) + 64,
                       stream);

        lstm_rec_kernel<<<NWG, 256, WREC_LDS_BYTES, stream>>>(xw, W_hh[l],
                                                              hbuf, cnt, layer_out);

        layer_in  = layer_out;
        layer_out = (layer_out == hA) ? hB : hA;
        K = Hdim;
    }

    mlp_head_kernel<<<Tdim, 32, 0, stream>>>(layer_in, W1, b1, W2, b2, out);
}